// GIN_20856361189655
// MI455X (gfx1250) — compile-verified
//
#include <hip/hip_runtime.h>
#include <hip/hip_bf16.h>

// ---------------------------------------------------------------------------
// GIN forward for MI455X (gfx1250, wave32).
// GEMMs use exact-FP32 WMMA: V_WMMA_F32_16X16X4_F32 (16x16 tile, K-step 4).
//   - Each wave owns one N-tile: its 128x16 B panel lives in 64 VGPRs,
//     loaded from L2 once and reused across 4 M-strips.
//   - A panel staged in LDS with dword stride K+4 (== 4 mod 64 banks) so the
//     per-step A fragment is one conflict-free ds_load_b64.
// Edge aggregation: wave-per-edge float4 gather + global f32 atomics (L2-resident).
// BatchNorm (training mode): per-column sum/sumsq accumulated by the GEMM kernel.
// ---------------------------------------------------------------------------

typedef __attribute__((ext_vector_type(2))) float v2f;
typedef __attribute__((ext_vector_type(8))) float v8f;

#define N_NODES 50000
#define N_EDGES 600000
#define IN_F    128
#define HID     128
#define NCLS    16
#define BN_EPS  1e-5f
#define BM      64      // rows of A per block in the GEMM

// ---------------------------------------------------------------- utilities
__global__ void zero_kernel(float* __restrict__ p, int n) {
    int i = blockIdx.x * blockDim.x + threadIdx.x;
    if (i < n) p[i] = 0.0f;
}

__global__ void copy4_kernel(float* __restrict__ dst, const float* __restrict__ src, int n4) {
    int i = blockIdx.x * blockDim.x + threadIdx.x;
    if (i < n4) ((float4*)dst)[i] = ((const float4*)src)[i];
}

// agg[dst[e]] += h[src[e]]  — one wave per edge, lane handles 4 contiguous floats.
__global__ void edge_scatter_kernel(float* __restrict__ agg, const float* __restrict__ h,
                                    const int* __restrict__ src, const int* __restrict__ dst,
                                    int nedges) {
    int gw   = (blockIdx.x * blockDim.x + threadIdx.x) >> 5;  // global wave id == edge id
    int lane = threadIdx.x & 31;
    if (gw >= nedges) return;
    int s = src[gw];
    int d = dst[gw];
    const float4 v = *(const float4*)(h + (size_t)s * 128 + lane * 4);
    float* p = agg + (size_t)d * 128 + lane * 4;
    atomicAdd(p + 0, v.x);
    atomicAdd(p + 1, v.y);
    atomicAdd(p + 2, v.z);
    atomicAdd(p + 3, v.w);
}

// ---------------------------------------------------------------- WMMA GEMM
// C[nrows x KOUT] = A[nrows x 128] @ W[128 x KOUT] + bias, plus per-column
// sum / sum-of-squares accumulation (for training-mode BatchNorm).
template<int KOUT>
__global__ void gemm_wmma_kernel(const float* __restrict__ A,
                                 const float* __restrict__ W,
                                 const float* __restrict__ bias,
                                 float* __restrict__ C,
                                 float* __restrict__ ssum,
                                 float* __restrict__ ssq,
                                 int nrows) {
    constexpr int K   = 128;
    constexpr int LDA = K + 4;                 // dword stride 132 == 4 (mod 64 banks)
    __shared__ float sA[BM][LDA];

    const int tid     = threadIdx.x;           // 0..255 (8 waves)
    const int lane    = tid & 31;
    const int wave    = tid >> 5;
    const int rowbase = blockIdx.x * BM;

    // Stage 64x128 A panel into LDS (float4, coalesced; OOB rows -> 0).
    for (int i = tid; i < BM * (K / 4); i += 256) {
        int r  = i / (K / 4);
        int c4 = (i % (K / 4)) * 4;
        float4 v = make_float4(0.f, 0.f, 0.f, 0.f);
        int gr = rowbase + r;
        if (gr < nrows) v = *(const float4*)(A + (size_t)gr * K + c4);
        sA[r][c4 + 0] = v.x; sA[r][c4 + 1] = v.y;
        sA[r][c4 + 2] = v.z; sA[r][c4 + 3] = v.w;
    }
    __syncthreads();

    const int l16   = lane & 15;
    const int hi    = lane >> 4;               // 0: lanes 0-15, 1: lanes 16-31
    const int khalf = hi * 2;                  // K offset of this lane half

    constexpr int NTILES = KOUT / 16;          // 8 or 1
    if (wave / NTILES < 4) {                   // wave-uniform: idle waves skip all
        const int n0 = (wave % NTILES) * 16;   // this wave's fixed N-tile

        // Preload the wave's 128x16 B panel into 64 VGPRs (32 k-steps x v2f).
        v2f breg[32];
        #pragma unroll
        for (int j = 0; j < 32; ++j) {
            const float* wp = W + (size_t)(4 * j + khalf) * KOUT + n0 + l16;
            breg[j].x = wp[0];
            breg[j].y = wp[KOUT];
        }
        const float bv = bias[n0 + l16];

        // Iterate M-strips, reusing the register-resident B panel.
        for (int ms = wave / NTILES; ms < 4; ms += 8 / NTILES) {
            const int mrow = ms * 16;
            v8f acc = { bv, bv, bv, bv, bv, bv, bv, bv };

            #pragma unroll
            for (int j = 0; j < 32; ++j) {
                // A 16x4 frag: one 8B LDS read per step (conflict-free banks)
                v2f a = *(const v2f*)&sA[mrow + l16][4 * j + khalf];
                acc = __builtin_amdgcn_wmma_f32_16x16x4_f32(
                    false, a, false, breg[j], (short)0, acc, false, false);
            }

            // Store D (VGPR i -> row i + 8*hi) and accumulate BN statistics.
            float s = 0.f, sq = 0.f;
            #pragma unroll
            for (int i = 0; i < 8; ++i) {
                int grow = rowbase + mrow + i + 8 * hi;
                float v  = acc[i];
                if (grow < nrows) {
                    C[(size_t)grow * KOUT + n0 + l16] = v;
                    s  += v;
                    sq += v * v;
                }
            }
            atomicAdd(&ssum[n0 + l16], s);
            atomicAdd(&ssq [n0 + l16], sq);
        }
    }
}

// ---------------------------------------------------------------- BN (+ReLU)
template<bool RELU>
__global__ void bn_act_kernel(float* __restrict__ X,
                              const float* __restrict__ ssum, const float* __restrict__ ssq,
                              const float* __restrict__ g,    const float* __restrict__ be,
                              int nrows, int F) {
    int idx = blockIdx.x * blockDim.x + threadIdx.x;
    int total = nrows * F;
    if (idx >= total) return;
    int c = idx & (F - 1);                      // F is a power of two (128)
    float invN = 1.0f / (float)nrows;
    float m    = ssum[c] * invN;
    float var  = fmaxf(ssq[c] * invN - m * m, 0.f);
    float sc   = g[c] * rsqrtf(var + BN_EPS);
    float sh   = be[c] - m * sc;
    float v    = X[idx] * sc + sh;
    if (RELU) v = fmaxf(v, 0.f);
    X[idx] = v;
}

// Fused: apply BN to the [nrows x 16] logits and accumulate column sums.
__global__ void final_reduce_kernel(const float* __restrict__ T,
                                    const float* __restrict__ ssum, const float* __restrict__ ssq,
                                    const float* __restrict__ g,    const float* __restrict__ be,
                                    float* __restrict__ acc, int nrows) {
    __shared__ float part[256];
    int tid = threadIdx.x;
    int col = tid & 15;
    float invN = 1.0f / (float)nrows;
    float m    = ssum[col] * invN;
    float var  = fmaxf(ssq[col] * invN - m * m, 0.f);
    float sc   = g[col] * rsqrtf(var + BN_EPS);
    float sh   = be[col] - m * sc;
    float s = 0.f;
    int total = nrows * 16;
    for (int idx = blockIdx.x * 256 + tid; idx < total; idx += gridDim.x * 256)
        s += T[idx] * sc + sh;                  // idx & 15 == col (stride is mult of 16)
    part[tid] = s;
    __syncthreads();
    if (tid < 16) {
        float tot = 0.f;
        #pragma unroll
        for (int i = 0; i < 256; i += 16) tot += part[tid + i];
        atomicAdd(&acc[tid], tot);
    }
}

__global__ void final_write_kernel(const float* __restrict__ acc, float* __restrict__ out,
                                   int nrows) {
    int c = threadIdx.x;
    if (c < NCLS) out[c] = acc[c] / (float)nrows;
}

// ---------------------------------------------------------------- launcher
extern "C" void kernel_launch(void* const* d_in, const int* in_sizes, int n_in,
                              void* d_out, int out_size, void* d_ws, size_t ws_size,
                              hipStream_t stream) {
    const float* x    = (const float*)d_in[0];
    const int*   src  = (const int*)  d_in[1];
    const int*   dstv = (const int*)  d_in[2];
    const float* w11  = (const float*)d_in[3];
    const float* b11  = (const float*)d_in[4];
    const float* g11  = (const float*)d_in[5];
    const float* be11 = (const float*)d_in[6];
    const float* w12  = (const float*)d_in[7];
    const float* b12  = (const float*)d_in[8];
    const float* g12  = (const float*)d_in[9];
    const float* be12 = (const float*)d_in[10];
    const float* w21  = (const float*)d_in[11];
    const float* b21  = (const float*)d_in[12];
    const float* g21  = (const float*)d_in[13];
    const float* be21 = (const float*)d_in[14];
    const float* w22  = (const float*)d_in[15];
    const float* b22  = (const float*)d_in[16];
    const float* g22  = (const float*)d_in[17];
    const float* be22 = (const float*)d_in[18];
    float* out = (float*)d_out;

    // Workspace layout (floats)
    float* bufA = (float*)d_ws;                         // [N x 128]
    float* bufB = bufA + (size_t)N_NODES * HID;         // [N x 128]
    float* bufC = bufB + (size_t)N_NODES * HID;         // [N x 16]
    float* st   = bufC + (size_t)N_NODES * NCLS;        // stats + out accumulator
    float *sum1 = st,       *sq1 = st + 128;
    float *sum2 = st + 256, *sq2 = st + 384;
    float *sum3 = st + 512, *sq3 = st + 640;
    float *sum4 = st + 768, *sq4 = st + 784;
    float *oacc = st + 800;                             // 16 floats (total 816)

    const int NF      = N_NODES * HID;                  // 6.4M
    const int gElem   = (NF + 255) / 256;
    const int gCopy   = (NF / 4 + 255) / 256;
    const int gEdge   = (N_EDGES * 32 + 255) / 256;     // wave-per-edge
    const int gGemm   = (N_NODES + BM - 1) / BM;

    // Deterministic: zero stat accumulators every call.
    zero_kernel<<<4, 256, 0, stream>>>(st, 816);

    // ---- layer 1: agg = x + sum_in(x) ----
    copy4_kernel<<<gCopy, 256, 0, stream>>>(bufA, x, NF / 4);
    edge_scatter_kernel<<<gEdge, 256, 0, stream>>>(bufA, x, src, dstv, N_EDGES);

    // ---- MLP 1 ----
    gemm_wmma_kernel<HID><<<gGemm, 256, 0, stream>>>(bufA, w11, b11, bufB, sum1, sq1, N_NODES);
    bn_act_kernel<true><<<gElem, 256, 0, stream>>>(bufB, sum1, sq1, g11, be11, N_NODES, HID);
    gemm_wmma_kernel<HID><<<gGemm, 256, 0, stream>>>(bufB, w12, b12, bufA, sum2, sq2, N_NODES);
    bn_act_kernel<true><<<gElem, 256, 0, stream>>>(bufA, sum2, sq2, g12, be12, N_NODES, HID);
    // (inner BN had no ReLU, but reference applies ReLU right after MLP1 -> fused)

    // ---- layer 2: agg = h + sum_in(h) ----
    copy4_kernel<<<gCopy, 256, 0, stream>>>(bufB, bufA, NF / 4);
    edge_scatter_kernel<<<gEdge, 256, 0, stream>>>(bufB, bufA, src, dstv, N_EDGES);

    // ---- MLP 2 ----
    gemm_wmma_kernel<HID><<<gGemm, 256, 0, stream>>>(bufB, w21, b21, bufA, sum3, sq3, N_NODES);
    bn_act_kernel<true><<<gElem, 256, 0, stream>>>(bufA, sum3, sq3, g21, be21, N_NODES, HID);
    gemm_wmma_kernel<NCLS><<<gGemm, 256, 0, stream>>>(bufA, w22, b22, bufC, sum4, sq4, N_NODES);

    // ---- final BN + mean over nodes ----
    final_reduce_kernel<<<3125, 256, 0, stream>>>(bufC, sum4, sq4, g22, be22, oacc, N_NODES);
    final_write_kernel<<<1, 32, 0, stream>>>(oacc, out, N_NODES);
}